// EnhancedMSTSN_7619271983414
// MI455X (gfx1250) — compile-verified
//
#include <hip/hip_runtime.h>
#include <hip/hip_bf16.h>
#include <math.h>

typedef __attribute__((ext_vector_type(2))) float v2f;
typedef __attribute__((ext_vector_type(8))) float v8f;

#define N_NODES 500
#define CAP     512
#define NGRAPH  96      // B*T
#define NROWS   48000   // NGRAPH*N_NODES == NSEQ*SLEN
#define NSEQ    2000    // B*N
#define SLEN    24

__device__ __forceinline__ float gelu_exact(float x) {
    return 0.5f * x * (1.0f + erff(x * 0.70710678118654752440f));
}

// ---------------- adjacency: cosine mask + deterministic CSR ----------------
__global__ void k_norms(const float* __restrict__ emb, float* __restrict__ rs) {
    int i = blockIdx.x * blockDim.x + threadIdx.x;
    if (i >= N_NODES) return;
    float s = 0.f;
#pragma unroll
    for (int d = 0; d < 8; ++d) { float e = emb[i * 8 + d]; s += e * e; }
    rs[i] = rsqrtf(s);
}

__global__ void k_mask(const float* __restrict__ emb, const float* __restrict__ rs,
                       unsigned char* __restrict__ mask) {
    int gid = blockIdx.x * blockDim.x + threadIdx.x;
    if (gid >= N_NODES * N_NODES) return;
    int i = gid / N_NODES, j = gid % N_NODES;
    float d = 0.f;
#pragma unroll
    for (int e = 0; e < 8; ++e) d += emb[i * 8 + e] * emb[j * 8 + e];
    mask[gid] = (d * rs[i] * rs[j] > 0.5f) ? 1 : 0;
}

__global__ void k_csr(const unsigned char* __restrict__ mask,
                      int* __restrict__ idx, int* __restrict__ cnt) {
    int i = blockIdx.x * blockDim.x + threadIdx.x;
    if (i >= N_NODES) return;
    int c = 0;
    for (int j = 0; j < N_NODES; ++j)
        if (mask[i * N_NODES + j]) idx[i * CAP + (c++)] = j;
    cnt[i] = c;   // always >=1 (self loop: cos=1)
}

// ---------------- f32 WMMA GEMM: C[M,NC] = A[M,KA] @ W[KA,NC] (+bias, +gelu) ----------------
// One wave computes one 16x16 output tile using chained V_WMMA_F32_16X16X4_F32.
template <int KA, int NC, int ACT>
__global__ void __launch_bounds__(128)
k_gemm(const float* __restrict__ A, const float* __restrict__ W,
       const float* __restrict__ bias, float* __restrict__ C, int M) {
    constexpr int KP = (KA + 3) & ~3;
    int mtiles = M >> 4;
    int mtile = blockIdx.x * 4 + (threadIdx.x >> 5);
    if (mtile >= mtiles) return;               // wave-uniform: EXEC stays all-1s
    int ntile = blockIdx.y;
    int lane  = threadIdx.x & 31;
    int n     = lane & 15;
    int ncol  = ntile * 16 + n;
    int khalf = (lane >> 4) * 2;               // 0 or 2
    int row   = mtile * 16 + n;                // A-fragment row for this lane
    v8f acc = {};
#pragma unroll
    for (int kb = 0; kb < KP; kb += 4) {
        int k0 = kb + khalf;
        v2f a, b;
        a.x = (k0     < KA) ? A[(size_t)row * KA + k0]     : 0.f;
        a.y = (k0 + 1 < KA) ? A[(size_t)row * KA + k0 + 1] : 0.f;
        bool nok = (ncol < NC);
        b.x = (nok && k0     < KA) ? W[(size_t)k0 * NC + ncol]       : 0.f;
        b.y = (nok && k0 + 1 < KA) ? W[(size_t)(k0 + 1) * NC + ncol] : 0.f;
        acc = __builtin_amdgcn_wmma_f32_16x16x4_f32(
                  false, a, false, b, (short)0, acc, false, false);
    }
    if (ncol < NC) {
        float bv = bias ? bias[ncol] : 0.f;
        int rbase = mtile * 16 + ((lane >> 4) * 8);
#pragma unroll
        for (int r = 0; r < 8; ++r) {
            float v = acc[r] + bv;
            if (ACT) v = gelu_exact(v);
            C[(size_t)(rbase + r) * NC + ncol] = v;
        }
    }
}

// ---------------- GAT per-node attention scores ----------------
template <int CC>
__global__ void k_gat_scores(const float* __restrict__ feat,
                             const float* __restrict__ a_src,
                             const float* __restrict__ a_dst,
                             float* __restrict__ ssrc, float* __restrict__ sdst) {
    int gid = blockIdx.x * blockDim.x + threadIdx.x;
    if (gid >= NROWS * 4) return;
    int nrow = gid >> 2, h = gid & 3;
    float s1 = 0.f, s2 = 0.f;
#pragma unroll
    for (int c = 0; c < CC; ++c) {
        float f = feat[(size_t)nrow * (4 * CC) + h * CC + c];
        s1 += f * a_src[h * CC + c];
        s2 += f * a_dst[h * CC + c];
    }
    ssrc[gid] = s1; sdst[gid] = s2;
}

// ---------------- GAT aggregation: masked edge-softmax + message sum ----------------
// Block = one graph (bt). Features + src-scores staged in LDS. Wave per row i.
template <int CC, int ACT>
__global__ void __launch_bounds__(512)
k_gat_agg(const float* __restrict__ feat, const float* __restrict__ ssrc,
          const float* __restrict__ sdst, const int* __restrict__ nidx,
          const int* __restrict__ ncnt, const float* __restrict__ bias,
          float* __restrict__ out) {
    constexpr int HC = 4 * CC;
    __shared__ float sfeat[N_NODES * HC];
    __shared__ float ss[N_NODES * 4];
    int bt = blockIdx.x;
    const float* fb = feat + (size_t)bt * N_NODES * HC;
    const float* sb = ssrc + (size_t)bt * N_NODES * 4;
    for (int i = threadIdx.x; i < N_NODES * HC; i += blockDim.x) sfeat[i] = fb[i];
    for (int i = threadIdx.x; i < N_NODES * 4;  i += blockDim.x) ss[i] = sb[i];
    __syncthreads();
    int lane = threadIdx.x & 31, wid = threadIdx.x >> 5, nw = blockDim.x >> 5;
    for (int i = wid; i < N_NODES; i += nw) {
        int row = bt * N_NODES + i;
        float sd[4];
#pragma unroll
        for (int h = 0; h < 4; ++h) sd[h] = sdst[(size_t)row * 4 + h];
        int cnt = ncnt[i];
        const int* lst = nidx + (size_t)i * CAP;
        // pass 1: per-head max of leaky_relu(sd+ss) over neighbors
        float mh[4];
#pragma unroll
        for (int h = 0; h < 4; ++h) mh[h] = -3.0e38f;
        for (int e = lane; e < cnt; e += 32) {
            int j = lst[e];
#pragma unroll
            for (int h = 0; h < 4; ++h) {
                float v = sd[h] + ss[j * 4 + h];
                v = (v > 0.f) ? v : 0.2f * v;
                mh[h] = fmaxf(mh[h], v);
            }
        }
#pragma unroll
        for (int off = 16; off > 0; off >>= 1)
#pragma unroll
            for (int h = 0; h < 4; ++h)
                mh[h] = fmaxf(mh[h], __shfl_xor(mh[h], off, 32));
        // pass 2: exp-weights, denominator, weighted message accumulation
        float den[4] = {0.f, 0.f, 0.f, 0.f};
        float num[HC];
#pragma unroll
        for (int ch = 0; ch < HC; ++ch) num[ch] = 0.f;
        for (int e = lane; e < cnt; e += 32) {
            int j = lst[e];
#pragma unroll
            for (int h = 0; h < 4; ++h) {
                float v = sd[h] + ss[j * 4 + h];
                v = (v > 0.f) ? v : 0.2f * v;
                float w = __expf(v - mh[h]);
                den[h] += w;
#pragma unroll
                for (int c = 0; c < CC; ++c)
                    num[h * CC + c] += w * sfeat[j * HC + h * CC + c];
            }
        }
#pragma unroll
        for (int off = 16; off > 0; off >>= 1) {
#pragma unroll
            for (int h = 0; h < 4; ++h) den[h] += __shfl_xor(den[h], off, 32);
#pragma unroll
            for (int ch = 0; ch < HC; ++ch) num[ch] += __shfl_xor(num[ch], off, 32);
        }
        if (lane == 0) {
#pragma unroll
            for (int h = 0; h < 4; ++h)
#pragma unroll
                for (int c = 0; c < CC; ++c) {
                    float v = num[h * CC + c] / den[h] + bias[h * CC + c];
                    if (ACT) v = gelu_exact(v);
                    out[(size_t)row * HC + h * CC + c] = v;
                }
        }
    }
}

// ---------------- MHA core: thread per (seq, head, query-pos) ----------------
__global__ void k_attn(const float* __restrict__ q, const float* __restrict__ k,
                       const float* __restrict__ v, float* __restrict__ o) {
    int gid = blockIdx.x * blockDim.x + threadIdx.x;
    if (gid >= NSEQ * 2 * SLEN) return;
    int s = gid % SLEN;
    int h = (gid / SLEN) & 1;
    int p = gid / (2 * SLEN);
    const float* qp = q + ((size_t)(p * SLEN + s)) * 16 + h * 8;
    float qr[8];
#pragma unroll
    for (int e = 0; e < 8; ++e) qr[e] = qp[e];
    float sc[SLEN];
    float mx = -3.0e38f;
    const float scale = 0.35355339059327376220f;  // 1/sqrt(8)
#pragma unroll
    for (int t = 0; t < SLEN; ++t) {
        const float* kp = k + ((size_t)(p * SLEN + t)) * 16 + h * 8;
        float d = 0.f;
#pragma unroll
        for (int e = 0; e < 8; ++e) d += qr[e] * kp[e];
        d *= scale;
        sc[t] = d;
        mx = fmaxf(mx, d);
    }
    float den = 0.f;
#pragma unroll
    for (int t = 0; t < SLEN; ++t) { float w = __expf(sc[t] - mx); sc[t] = w; den += w; }
    float inv = 1.f / den;
    float acc[8];
#pragma unroll
    for (int e = 0; e < 8; ++e) acc[e] = 0.f;
#pragma unroll
    for (int t = 0; t < SLEN; ++t) {
        const float* vp = v + ((size_t)(p * SLEN + t)) * 16 + h * 8;
        float a = sc[t] * inv;
#pragma unroll
        for (int e = 0; e < 8; ++e) acc[e] += a * vp[e];
    }
    float* op = o + ((size_t)(p * SLEN + s)) * 16 + h * 8;
#pragma unroll
    for (int e = 0; e < 8; ++e) op[e] = acc[e];
}

// ---------------- residual + LayerNorm (eps=1e-3, biased var) ----------------
__global__ void k_add_ln(const float* __restrict__ x, const float* __restrict__ r,
                         const float* __restrict__ g, const float* __restrict__ b,
                         float* __restrict__ out, int M) {
    int row = blockIdx.x * blockDim.x + threadIdx.x;
    if (row >= M) return;
    float v[16];
    float m = 0.f;
#pragma unroll
    for (int d = 0; d < 16; ++d) {
        float t = x[(size_t)row * 16 + d] + r[(size_t)row * 16 + d];
        v[d] = t; m += t;
    }
    m *= (1.f / 16.f);
    float var = 0.f;
#pragma unroll
    for (int d = 0; d < 16; ++d) { float t = v[d] - m; var += t * t; }
    var *= (1.f / 16.f);
    float is = rsqrtf(var + 1e-3f);
#pragma unroll
    for (int d = 0; d < 16; ++d)
        out[(size_t)row * 16 + d] = (v[d] - m) * is * g[d] + b[d];
}

// ---------------- mean over time ----------------
__global__ void k_pool(const float* __restrict__ t, float* __restrict__ pooled) {
    int gid = blockIdx.x * blockDim.x + threadIdx.x;
    if (gid >= NSEQ * 16) return;
    int p = gid >> 4, d = gid & 15;
    float s = 0.f;
#pragma unroll
    for (int ss = 0; ss < SLEN; ++ss) s += t[((size_t)p * SLEN + ss) * 16 + d];
    pooled[gid] = s * (1.f / 24.f);
}

// ---------------- final regressor dot ----------------
__global__ void k_final(const float* __restrict__ g, const float* __restrict__ w2,
                        const float* __restrict__ b2, float* __restrict__ out) {
    int p = blockIdx.x * blockDim.x + threadIdx.x;
    if (p >= NSEQ) return;
    float s = 0.f;
#pragma unroll
    for (int d = 0; d < 16; ++d) s += g[(size_t)p * 16 + d] * w2[d];
    out[p] = s + b2[0];
}

extern "C" void kernel_launch(void* const* d_in, const int* in_sizes, int n_in,
                              void* d_out, int out_size, void* d_ws, size_t ws_size,
                              hipStream_t stream) {
    (void)in_sizes; (void)n_in; (void)out_size; (void)ws_size;
    const float* x      = (const float*)d_in[0];
    const float* emb    = (const float*)d_in[1];
    const float* proj_w = (const float*)d_in[2];
    const float* proj_b = (const float*)d_in[3];
    const float* g1_w   = (const float*)d_in[4];
    const float* g1_as  = (const float*)d_in[5];
    const float* g1_ad  = (const float*)d_in[6];
    const float* g1_b   = (const float*)d_in[7];
    const float* g2_w   = (const float*)d_in[8];
    const float* g2_as  = (const float*)d_in[9];
    const float* g2_ad  = (const float*)d_in[10];
    const float* g2_b   = (const float*)d_in[11];
    const float* wq     = (const float*)d_in[12];
    const float* bq     = (const float*)d_in[13];
    const float* wk     = (const float*)d_in[14];
    const float* bk     = (const float*)d_in[15];
    const float* wv     = (const float*)d_in[16];
    const float* bv     = (const float*)d_in[17];
    const float* wo     = (const float*)d_in[18];
    const float* bo     = (const float*)d_in[19];
    const float* ln1_g  = (const float*)d_in[20];
    const float* ln1_b  = (const float*)d_in[21];
    const float* ln2_g  = (const float*)d_in[22];
    const float* ln2_b  = (const float*)d_in[23];
    const float* f_w1   = (const float*)d_in[24];
    const float* f_b1   = (const float*)d_in[25];
    const float* f_w2   = (const float*)d_in[26];
    const float* f_b2   = (const float*)d_in[27];
    const float* r_w1   = (const float*)d_in[28];
    const float* r_b1   = (const float*)d_in[29];
    const float* r_w2   = (const float*)d_in[30];
    const float* r_b2   = (const float*)d_in[31];
    float* out = (float*)d_out;

    char* ws = (char*)d_ws;
    size_t off = 0;
    auto alloc = [&](size_t bytes) -> char* {
        char* p = ws + off;
        off += (bytes + 255) & ~(size_t)255;
        return p;
    };
    float* rs           = (float*)alloc(N_NODES * 4);
    unsigned char* mask = (unsigned char*)alloc(N_NODES * N_NODES);
    int*   ncnt         = (int*)alloc(N_NODES * 4);
    int*   nidx         = (int*)alloc((size_t)N_NODES * CAP * 4);
    float* h0    = (float*)alloc((size_t)NROWS * 8 * 4);
    float* feat1 = (float*)alloc((size_t)NROWS * 8 * 4);
    float* ssrc1 = (float*)alloc((size_t)NROWS * 4 * 4);
    float* sdst1 = (float*)alloc((size_t)NROWS * 4 * 4);
    float* h1    = (float*)alloc((size_t)NROWS * 8 * 4);
    float* feat2 = (float*)alloc((size_t)NROWS * 16 * 4);
    float* ssrc2 = (float*)alloc((size_t)NROWS * 4 * 4);
    float* sdst2 = (float*)alloc((size_t)NROWS * 4 * 4);
    float* h2    = (float*)alloc((size_t)NROWS * 16 * 4);
    float* qb    = (float*)alloc((size_t)NROWS * 16 * 4);
    float* kb    = (float*)alloc((size_t)NROWS * 16 * 4);
    float* vb    = (float*)alloc((size_t)NROWS * 16 * 4);
    float* ob    = (float*)alloc((size_t)NROWS * 16 * 4);
    float* ao    = (float*)alloc((size_t)NROWS * 16 * 4);
    float* t1    = (float*)alloc((size_t)NROWS * 16 * 4);
    float* ffh   = (float*)alloc((size_t)NROWS * 32 * 4);
    float* ffo   = (float*)alloc((size_t)NROWS * 16 * 4);
    float* t2    = (float*)alloc((size_t)NROWS * 16 * 4);
    float* pooled= (float*)alloc((size_t)NSEQ * 16 * 4);
    float* regh  = (float*)alloc((size_t)NSEQ * 16 * 4);

    // adjacency (batch-invariant, computed once per call)
    k_norms<<<(N_NODES + 255) / 256, 256, 0, stream>>>(emb, rs);
    k_mask<<<(N_NODES * N_NODES + 255) / 256, 256, 0, stream>>>(emb, rs, mask);
    k_csr<<<(N_NODES + 255) / 256, 256, 0, stream>>>(mask, nidx, ncnt);

    // spatial: proj -> GAT1(gelu) -> GAT2
    k_gemm<3, 8, 0><<<dim3(750, 1), 128, 0, stream>>>(x, proj_w, proj_b, h0, NROWS);
    k_gemm<8, 8, 0><<<dim3(750, 1), 128, 0, stream>>>(h0, g1_w, nullptr, feat1, NROWS);
    k_gat_scores<2><<<(NROWS * 4 + 255) / 256, 256, 0, stream>>>(feat1, g1_as, g1_ad, ssrc1, sdst1);
    k_gat_agg<2, 1><<<NGRAPH, 512, 0, stream>>>(feat1, ssrc1, sdst1, nidx, ncnt, g1_b, h1);
    k_gemm<8, 16, 0><<<dim3(750, 1), 128, 0, stream>>>(h1, g2_w, nullptr, feat2, NROWS);
    k_gat_scores<4><<<(NROWS * 4 + 255) / 256, 256, 0, stream>>>(feat2, g2_as, g2_ad, ssrc2, sdst2);
    k_gat_agg<4, 0><<<NGRAPH, 512, 0, stream>>>(feat2, ssrc2, sdst2, nidx, ncnt, g2_b, h2);

    // transformer (h2 buffer IS t[2000,24,16] — raw reshape is flat-identical)
    k_gemm<16, 16, 0><<<dim3(750, 1), 128, 0, stream>>>(h2, wq, bq, qb, NROWS);
    k_gemm<16, 16, 0><<<dim3(750, 1), 128, 0, stream>>>(h2, wk, bk, kb, NROWS);
    k_gemm<16, 16, 0><<<dim3(750, 1), 128, 0, stream>>>(h2, wv, bv, vb, NROWS);
    k_attn<<<(NSEQ * 2 * SLEN + 255) / 256, 256, 0, stream>>>(qb, kb, vb, ob);
    k_gemm<16, 16, 0><<<dim3(750, 1), 128, 0, stream>>>(ob, wo, bo, ao, NROWS);
    k_add_ln<<<(NROWS + 255) / 256, 256, 0, stream>>>(h2, ao, ln1_g, ln1_b, t1, NROWS);
    k_gemm<16, 32, 1><<<dim3(750, 2), 128, 0, stream>>>(t1, f_w1, f_b1, ffh, NROWS);
    k_gemm<32, 16, 0><<<dim3(750, 1), 128, 0, stream>>>(ffh, f_w2, f_b2, ffo, NROWS);
    k_add_ln<<<(NROWS + 255) / 256, 256, 0, stream>>>(t1, ffo, ln2_g, ln2_b, t2, NROWS);

    // pool + regressor
    k_pool<<<(NSEQ * 16 + 255) / 256, 256, 0, stream>>>(t2, pooled);
    k_gemm<16, 16, 1><<<dim3((125 + 3) / 4, 1), 128, 0, stream>>>(pooled, r_w1, r_b1, regh, NSEQ);
    k_final<<<(NSEQ + 255) / 256, 256, 0, stream>>>(regh, r_w2, r_b2, out);
}